// RGP_53300544143661
// MI455X (gfx1250) — compile-verified
//
#include <hip/hip_runtime.h>
#include <math.h>

typedef float v2f __attribute__((ext_vector_type(2)));
typedef float v8f __attribute__((ext_vector_type(8)));

#define QD     6
#define LTC    10
#define DD     60     // Lt*Q
#define KP     64     // DD padded to WMMA K multiple
#define MM     256    // inducing points
#define NSPLIT 16     // n-slices for acc partials
#define JITTER 1e-6f

// ---------------- stage 1: hankel-derived elementwise arrays (padded to KP, pads=0) ----
__global__ void prep_elems_kernel(const float* __restrict__ Xm, const float* __restrict__ Xv,
                                  const float* __restrict__ kl,
                                  float* __restrict__ mu, float* __restrict__ w,
                                  float* __restrict__ wm, float* __restrict__ sw,
                                  float* __restrict__ d1inv, int Nw)
{
  int idx = blockIdx.x * blockDim.x + threadIdx.x;
  if (idx >= Nw * KP) return;
  int n = idx >> 6, d = idx & 63;
  float l2 = kl[0] * kl[0];
  float muv = 0.f, wv = 0.f, wmv = 0.f, swv = 0.f, d1 = 0.f;
  if (d < DD) {
    int l = d / QD, q = d - l * QD;
    float m_ = Xm[(n + l) * QD + q];
    float S  = Xv[(n + l) * QD + q];
    muv = m_;
    wv  = 1.f / (2.f * S + l2);
    wmv = wv * m_;
    swv = sqrtf(wv);
    d1  = 1.f / (S + l2);
  }
  mu[idx] = muv; w[idx] = wv; wm[idx] = wmv; sw[idx] = swv; d1inv[idx] = d1;
}

// ---------------- stage 2: per-row reductions ld1[n], gf[n]=0.5*(ld2-e1) ----------------
__global__ void prep_rows_kernel(const float* __restrict__ Xm, const float* __restrict__ Xv,
                                 const float* __restrict__ kl,
                                 float* __restrict__ ld1, float* __restrict__ gf, int Nw)
{
  int n = blockIdx.x * blockDim.x + threadIdx.x;
  if (n >= Nw) return;
  float l2 = kl[0] * kl[0];
  float s1 = 0.f, s2 = 0.f, e1 = 0.f;
  for (int d = 0; d < DD; ++d) {
    int l = d / QD, q = d - l * QD;
    float m_ = Xm[(n + l) * QD + q];
    float S  = Xv[(n + l) * QD + q];
    s1 += log1pf(S / l2);
    s2 += log1pf(2.f * S / l2);
    float wv = 1.f / (2.f * S + l2);
    e1 += wv * m_ * m_;
  }
  ld1[n] = -0.5f * s1;
  gf[n]  = 0.5f * (-0.5f * s2 - e1);
}

// ---------------- Z padded to stride KP ----------------
__global__ void padz_kernel(const float* __restrict__ Z, float* __restrict__ Zp)
{
  int idx = blockIdx.x * blockDim.x + threadIdx.x;
  if (idx >= MM * KP) return;
  int m = idx >> 6, d = idx & 63;
  Zp[idx] = (d < DD) ? Z[m * DD + d] : 0.f;
}

// ---------------- zz and Kuu ----------------
__global__ void zz_kuu_kernel(const float* __restrict__ Zp, const float* __restrict__ kv,
                              const float* __restrict__ kl,
                              float* __restrict__ zz, float* __restrict__ Kuu)
{
  int idx = blockIdx.x * blockDim.x + threadIdx.x;
  if (idx >= MM * MM) return;
  int m = idx >> 8, p = idx & 255;
  const float* a = Zp + m * KP;
  const float* b = Zp + p * KP;
  float s = 0.f;
  for (int d = 0; d < DD; ++d) { float df = a[d] - b[d]; s += df * df; }
  zz[idx] = s;
  float l2 = kl[0] * kl[0];
  Kuu[idx] = kv[0] * expf(-0.5f * s / l2) + ((m == p) ? JITTER : 0.f);
}

// ---------------- f[n,m] = gf[n] + e2[n,m] - e3[n,m] ----------------
__global__ __launch_bounds__(256)
void f_kernel(const float* __restrict__ wm, const float* __restrict__ w,
              const float* __restrict__ gf, const float* __restrict__ Zp,
              float* __restrict__ f, int Nw)
{
  size_t idx = (size_t)blockIdx.x * blockDim.x + threadIdx.x;
  if (idx >= (size_t)Nw * MM) return;
  int n = (int)(idx >> 8), m = (int)(idx & 255);
  const float* wmr = wm + (size_t)n * KP;
  const float* wr  = w  + (size_t)n * KP;
  const float* zr  = Zp + m * KP;
  float s = 0.f;
  for (int d = 0; d < DD; ++d) {
    float z = zr[d];
    s += wmr[d] * z - 0.25f * wr[d] * z * z;
  }
  f[idx] = gf[n] + s;
}

// ---------------- acc kernel: per-n Gram via V_WMMA_F32_16X16X4_F32, exp, accumulate ----
// arg[n,m,p] = f[n,m] + f[n,p] - 0.5 * sum_d w[n,d] Z[m,d] Z[p,d]
// A = ((-0.5*Z_mrows) .* sqrt(w_n)), B = (Z_prows .* sqrt(w_n))  (Gram lane-symmetry).
// The -0.5 factor is pre-folded into the loop-invariant za registers so the per-n
// inner loop only does 2 packed multiplies per K-step.
__global__ __launch_bounds__(512)
void acc_kernel(const float* __restrict__ Zp, const float* __restrict__ sw,
                const float* __restrict__ f, float* __restrict__ accpart, int Nw)
{
  const int wid  = threadIdx.x >> 5;
  const int lane = threadIdx.x & 31;
  const int mpb = blockIdx.x & 15;   // which 64x64 (m,p) block
  const int nsl = blockIdx.x >> 4;   // which n-slice
  const int bm = mpb >> 2, bp = mpb & 3;
  const int m0 = bm * 64 + (wid >> 2) * 16;
  const int p0 = bp * 64 + (wid & 3) * 16;
  const int row  = lane & 15;
  const int koff = (lane >> 4) << 1; // K sub-offset per A/B layout (lanes16-31 hold K+2)
  const int hi8  = (lane >> 4) << 3; // C/D layout: lanes16-31 hold M+8

  // Preload static Z tiles in WMMA A/B register layout (K=64 -> 16 steps of K=4).
  // za carries the -0.5 Gram scale (loop-invariant fold).
  float za[32], zb[32];
#pragma unroll
  for (int s = 0; s < 16; ++s) {
    int d = 4 * s + koff;
    float2 a = *(const float2*)(Zp + (m0 + row) * KP + d);
    float2 b = *(const float2*)(Zp + (p0 + row) * KP + d);
    za[2 * s] = -0.5f * a.x; za[2 * s + 1] = -0.5f * a.y;
    zb[2 * s] = b.x;         zb[2 * s + 1] = b.y;
  }

  int nper = (Nw + NSPLIT - 1) / NSPLIT;
  int nbeg = nsl * nper;
  int nend = nbeg + nper; if (nend > Nw) nend = Nw;

  v8f accv = {0.f, 0.f, 0.f, 0.f, 0.f, 0.f, 0.f, 0.f};

  for (int n = nbeg; n < nend; ++n) {
    const float* swr = sw + (size_t)n * KP;
    v8f gram = {0.f, 0.f, 0.f, 0.f, 0.f, 0.f, 0.f, 0.f};
#pragma unroll
    for (int s = 0; s < 16; ++s) {
      int d = 4 * s + koff;
      float2 swv = *(const float2*)(swr + d);
      v2f A, B;
      A.x = za[2 * s]     * swv.x;
      A.y = za[2 * s + 1] * swv.y;
      B.x = zb[2 * s]     * swv.x;
      B.y = zb[2 * s + 1] * swv.y;
      gram = __builtin_amdgcn_wmma_f32_16x16x4_f32(false, A, false, B,
                                                   (short)0, gram, false, false);
    }
    const float* fr = f + (size_t)n * MM;
    float fp = fr[p0 + row];
#pragma unroll
    for (int r = 0; r < 8; ++r) {
      accv[r] += __expf(gram[r] + fr[m0 + r + hi8] + fp);
    }
  }

  float* dst = accpart + (size_t)nsl * MM * MM;
#pragma unroll
  for (int r = 0; r < 8; ++r) {
    int m = m0 + r + hi8;
    int p = p0 + row;
    dst[m * MM + p] = accv[r];  // disjoint per workgroup -> plain store, deterministic
  }
}

// ---------------- psi2 = sf2^2 * exp(-zz/(4 l2)) * sum_slices accpart ----------------
__global__ void psi2_kernel(const float* __restrict__ accpart, const float* __restrict__ zz,
                            const float* __restrict__ kv, const float* __restrict__ kl,
                            float* __restrict__ psi2)
{
  int idx = blockIdx.x * blockDim.x + threadIdx.x;
  if (idx >= MM * MM) return;
  float s = 0.f;
  for (int b = 0; b < NSPLIT; ++b) s += accpart[(size_t)b * MM * MM + idx];
  float l2 = kl[0] * kl[0];
  float sf2 = kv[0];
  psi2[idx] = sf2 * sf2 * __expf(-zz[idx] / (4.f * l2)) * s;
}

// ---------------- single-workgroup 256x256 Cholesky (in global, copy then factor) ------
__global__ __launch_bounds__(1024)
void chol_kernel(const float* __restrict__ src, float* __restrict__ A)
{
  int t = threadIdx.x, nt = blockDim.x;
  for (int i = t; i < MM * MM; i += nt) A[i] = src[i];
  __syncthreads();
  for (int k = 0; k < MM; ++k) {
    if (t == 0) A[k * MM + k] = sqrtf(A[k * MM + k]);
    __syncthreads();
    float dk = A[k * MM + k];
    for (int i = k + 1 + t; i < MM; i += nt) A[i * MM + k] /= dk;
    __syncthreads();
    int rem = MM - 1 - k;
    for (int idx = t; idx < rem * rem; idx += nt) {
      int i = k + 1 + idx / rem, j = k + 1 + idx % rem;
      if (j <= i) A[i * MM + j] -= A[i * MM + k] * A[j * MM + k];
    }
    __syncthreads();
  }
}

// ---------------- forward substitution: L X = B, wave-per-column, x in LDS -------------
__global__ __launch_bounds__(256)
void trisolve_kernel(const float* __restrict__ Lm, const float* __restrict__ Bin,
                     float* __restrict__ Xout, int ncols, int transB)
{
  __shared__ float xs[8][MM];
  int wid = threadIdx.x >> 5, lane = threadIdx.x & 31;
  int col = blockIdx.x * 8 + wid;
  bool act = col < ncols;
  if (act)
    for (int i = lane; i < MM; i += 32)
      xs[wid][i] = transB ? Bin[(size_t)col * MM + i] : Bin[(size_t)i * ncols + col];
  __syncthreads();
  for (int i = 0; i < MM; ++i) {
    float s = 0.f;
    if (act)
      for (int j = lane; j < i; j += 32) s += Lm[i * MM + j] * xs[wid][j];
    for (int o = 16; o > 0; o >>= 1) s += __shfl_down(s, o, 32);
    if (act && lane == 0) xs[wid][i] = (xs[wid][i] - s) / Lm[i * MM + i];
    __syncthreads();
  }
  if (act)
    for (int i = lane; i < MM; i += 32)
      Xout[(size_t)i * ncols + col] = xs[wid][i];
}

// ---------------- B = AATraw/sigma2 + I ----------------
__global__ void buildB_kernel(const float* __restrict__ AATraw, const float* __restrict__ lv,
                              float* __restrict__ B)
{
  int idx = blockIdx.x * blockDim.x + threadIdx.x;
  if (idx >= MM * MM) return;
  B[idx] = AATraw[idx] / lv[0] + (((idx >> 8) == (idx & 255)) ? 1.f : 0.f);
}

// ---------------- fused psi1 + Y = psi1^T X_mo (per-block partials) --------------------
__global__ __launch_bounds__(256)
void psi1y_kernel(const float* __restrict__ mu, const float* __restrict__ d1inv,
                  const float* __restrict__ ld1, const float* __restrict__ Zp,
                  const float* __restrict__ Xm, const float* __restrict__ kv,
                  float* __restrict__ ypart, int Nw)
{
  __shared__ float smu[KP], sd1[KP], sxmo[QD];
  int t = threadIdx.x;   // t == m
  float sf2 = kv[0];
  float yacc[QD] = {0.f, 0.f, 0.f, 0.f, 0.f, 0.f};
  int n0 = blockIdx.x * 32;
  for (int n = n0; n < n0 + 32 && n < Nw; ++n) {
    __syncthreads();
    if (t < KP) { smu[t] = mu[(size_t)n * KP + t]; sd1[t] = d1inv[(size_t)n * KP + t]; }
    if (t < QD) sxmo[t] = Xm[(n + LTC) * QD + t];
    __syncthreads();
    float s = 0.f;
    const float* zr = Zp + t * KP;
    for (int d = 0; d < DD; ++d) {
      float diff = smu[d] - zr[d];
      s += diff * diff * sd1[d];
    }
    float p1 = sf2 * __expf(ld1[n] - 0.5f * s);
#pragma unroll
    for (int q = 0; q < QD; ++q) yacc[q] += p1 * sxmo[q];
  }
  float* yp = ypart + (size_t)blockIdx.x * MM * QD;
#pragma unroll
  for (int q = 0; q < QD; ++q) yp[t * QD + q] = yacc[q];
}

__global__ void ysum_kernel(const float* __restrict__ ypart, float* __restrict__ Y, int nb)
{
  int idx = blockIdx.x * blockDim.x + threadIdx.x;
  if (idx >= MM * QD) return;
  float s = 0.f;
  for (int b = 0; b < nb; ++b) s += ypart[(size_t)b * MM * QD + idx];
  Y[idx] = s;
}

// ---------------- observation sums: sum X_vo, sum X_mo^2, sum log X_vo ------------------
__global__ __launch_bounds__(256)
void reduce_obs_kernel(const float* __restrict__ Xm, const float* __restrict__ Xv,
                       float* __restrict__ obspart, int Nw)
{
  __shared__ float r0[256], r1[256], r2[256];
  int t = threadIdx.x;
  float a = 0.f, b = 0.f, c = 0.f;
  int total = Nw * QD;
  for (int idx = blockIdx.x * 256 + t; idx < total; idx += gridDim.x * 256) {
    float m_ = Xm[idx + LTC * QD];
    float v  = Xv[idx + LTC * QD];
    a += v; b += m_ * m_; c += logf(v);
  }
  r0[t] = a; r1[t] = b; r2[t] = c;
  __syncthreads();
  for (int s = 128; s > 0; s >>= 1) {
    if (t < s) { r0[t] += r0[t + s]; r1[t] += r1[t + s]; r2[t] += r2[t + s]; }
    __syncthreads();
  }
  if (t == 0) {
    obspart[blockIdx.x * 3 + 0] = r0[0];
    obspart[blockIdx.x * 3 + 1] = r1[0];
    obspart[blockIdx.x * 3 + 2] = r2[0];
  }
}

// ---------------- final scalar assembly (double precision combine) ---------------------
__global__ void final_kernel(const float* __restrict__ AATraw, const float* __restrict__ LB,
                             const float* __restrict__ Cc, const float* __restrict__ obspart,
                             const float* __restrict__ Xm, const float* __restrict__ Xv,
                             const float* __restrict__ kv, const float* __restrict__ kl,
                             const float* __restrict__ lv, float* __restrict__ out,
                             int Nw, int nObsBlocks)
{
  double sigma2 = (double)lv[0];
  double sf2 = (double)kv[0];
  double sv = 0, sm2 = 0, slogv = 0;
  for (int b = 0; b < nObsBlocks; ++b) {
    sv    += obspart[b * 3 + 0];
    sm2   += obspart[b * 3 + 1];
    slogv += obspart[b * 3 + 2];
  }
  double trA = 0;
  for (int i = 0; i < MM; ++i) trA += AATraw[i * MM + i];
  trA /= sigma2;
  double ldB = 0;
  for (int i = 0; i < MM; ++i) ldB += log((double)LB[i * MM + i]);
  ldB *= 2.0;
  double sc2 = 0;
  for (int i = 0; i < MM * QD; ++i) { double c = Cc[i]; sc2 += c * c; }
  sc2 /= sigma2 * sigma2;
  double sb = 0;
  for (int i = 0; i < LTC * QD; ++i) { double m = Xm[i], v = Xv[i]; sb += m * m + v; }
  const double two_pi = 6.283185307179586476925287;
  double Nwd = (double)Nw, Qdd = (double)QD, Ltd = (double)LTC;
  double psi0 = Nwd * sf2;
  double ent  = 0.5 * slogv + Nwd * Qdd * 0.5 * log(two_pi);
  double ent2 = -Ltd * Qdd * log(two_pi) - 0.5 * sb;
  double bound = -0.5 * Nwd * Qdd * log(two_pi * sigma2);
  bound += -0.5 / sigma2 * (sv + sm2);
  bound += -0.5 * Qdd * (psi0 / sigma2 - trA);
  bound += -0.5 * Qdd * ldB;
  bound += 0.5 * sc2;
  bound += ent + ent2;
  out[0] = (float)bound;
}

// =======================================================================================
extern "C" void kernel_launch(void* const* d_in, const int* in_sizes, int n_in,
                              void* d_out, int out_size, void* d_ws, size_t ws_size,
                              hipStream_t stream)
{
  const float* Xm = (const float*)d_in[0];
  const float* Xv = (const float*)d_in[1];
  const float* Z  = (const float*)d_in[2];
  const float* kv = (const float*)d_in[3];
  const float* kl = (const float*)d_in[4];
  const float* lv = (const float*)d_in[5];

  int Nt = in_sizes[0] / QD;   // 8202
  int Nw = Nt - LTC;           // 8192 (hankel rows)

  float* ws = (float*)d_ws;
  size_t off = 0;
  auto alloc = [&](size_t n) { float* p = ws + off; off += n; return p; };

  float* mu      = alloc((size_t)Nw * KP);
  float* w       = alloc((size_t)Nw * KP);
  float* wm      = alloc((size_t)Nw * KP);
  float* sw      = alloc((size_t)Nw * KP);
  float* d1inv   = alloc((size_t)Nw * KP);
  float* ld1     = alloc((size_t)Nw);
  float* gf      = alloc((size_t)Nw);
  float* Zp      = alloc((size_t)MM * KP);
  float* f       = alloc((size_t)Nw * MM);
  float* accpart = alloc((size_t)NSPLIT * MM * MM);
  float* zz      = alloc((size_t)MM * MM);
  float* Kuu     = alloc((size_t)MM * MM);
  float* Lc      = alloc((size_t)MM * MM);
  float* psi2    = alloc((size_t)MM * MM);
  float* tmp     = alloc((size_t)MM * MM);
  float* AATraw  = alloc((size_t)MM * MM);
  float* Bm      = alloc((size_t)MM * MM);
  float* LBc     = alloc((size_t)MM * MM);
  int NB1 = (Nw + 31) / 32;
  float* ypart   = alloc((size_t)NB1 * MM * QD);
  float* Y       = alloc((size_t)MM * QD);
  float* Y2      = alloc((size_t)MM * QD);
  float* Cc      = alloc((size_t)MM * QD);
  float* obspart = alloc((size_t)256 * 3);

  prep_elems_kernel<<<(Nw * KP + 255) / 256, 256, 0, stream>>>(Xm, Xv, kl, mu, w, wm, sw, d1inv, Nw);
  prep_rows_kernel<<<(Nw + 255) / 256, 256, 0, stream>>>(Xm, Xv, kl, ld1, gf, Nw);
  padz_kernel<<<(MM * KP + 255) / 256, 256, 0, stream>>>(Z, Zp);
  zz_kuu_kernel<<<(MM * MM + 255) / 256, 256, 0, stream>>>(Zp, kv, kl, zz, Kuu);
  f_kernel<<<(unsigned)(((size_t)Nw * MM + 255) / 256), 256, 0, stream>>>(wm, w, gf, Zp, f, Nw);

  acc_kernel<<<16 * NSPLIT, 512, 0, stream>>>(Zp, sw, f, accpart, Nw);
  psi2_kernel<<<(MM * MM + 255) / 256, 256, 0, stream>>>(accpart, zz, kv, kl, psi2);

  chol_kernel<<<1, 1024, 0, stream>>>(Kuu, Lc);
  trisolve_kernel<<<(MM + 7) / 8, 256, 0, stream>>>(Lc, psi2, tmp, MM, 0);
  trisolve_kernel<<<(MM + 7) / 8, 256, 0, stream>>>(Lc, tmp, AATraw, MM, 1);
  buildB_kernel<<<(MM * MM + 255) / 256, 256, 0, stream>>>(AATraw, lv, Bm);
  chol_kernel<<<1, 1024, 0, stream>>>(Bm, LBc);

  psi1y_kernel<<<NB1, 256, 0, stream>>>(mu, d1inv, ld1, Zp, Xm, kv, ypart, Nw);
  ysum_kernel<<<(MM * QD + 255) / 256, 256, 0, stream>>>(ypart, Y, NB1);
  trisolve_kernel<<<1, 256, 0, stream>>>(Lc, Y, Y2, QD, 0);
  trisolve_kernel<<<1, 256, 0, stream>>>(LBc, Y2, Cc, QD, 0);

  reduce_obs_kernel<<<256, 256, 0, stream>>>(Xm, Xv, obspart, Nw);
  final_kernel<<<1, 1, 0, stream>>>(AATraw, LBc, Cc, obspart, Xm, Xv, kv, kl, lv,
                                    (float*)d_out, Nw, 256);
}